// GCN_79723182948631
// MI455X (gfx1250) — compile-verified
//
#include <hip/hip_runtime.h>

// GCN layer: segment-mean of gathered neighbor features, then ReLU(h @ W^T + b).
// Phase 1/2: L2-resident gather + f32 atomic scatter-add (wave-per-edge, float4 loads).
// Phase 3: full-precision WMMA GEMM via V_WMMA_F32_16X16X4_F32, mean fused into A load.

#define NNODES 50000
#define NEDGES 800000
#define DIM    128

typedef __attribute__((ext_vector_type(2))) float v2f;
typedef __attribute__((ext_vector_type(8))) float v8f;

// ---------------------------------------------------------------- zero scratch
__global__ void gcn_zero_ws(float* __restrict__ ws, int total) {
    int i = blockIdx.x * blockDim.x + threadIdx.x;
    if (i < total) ws[i] = 0.0f;
}

// ------------------------------------------------- edge scatter: sum + degree
// One wave (32 lanes) per edge. Each lane moves 4 consecutive floats (float4
// gather, 4x global_atomic_add_f32 scatter). Lane 0 bumps the in-degree count.
__global__ void gcn_scatter(const float* __restrict__ feat,
                            const int*   __restrict__ src,
                            const int*   __restrict__ dst,
                            float* __restrict__ sums,
                            float* __restrict__ cnt) {
    int edge = (blockIdx.x * blockDim.x + threadIdx.x) >> 5;
    int lane = threadIdx.x & 31;
    if (edge >= NEDGES) return;
    int s = src[edge];
    int d = dst[edge];
    const float4 v = *((const float4*)(feat + (long long)s * DIM) + lane);
    float* o = sums + (long long)d * DIM + lane * 4;
    atomicAdd(o + 0, v.x);
    atomicAdd(o + 1, v.y);
    atomicAdd(o + 2, v.z);
    atomicAdd(o + 3, v.w);
    if (lane == 0) atomicAdd(cnt + d, 1.0f);
}

// --------------------------------------- fused mean + GEMM + bias + ReLU (WMMA)
// out[m][n] = relu( (sums[m][:] / max(cnt[m],1)) . W[n][:] + b[n] )
// One wave computes one 16x16 output tile with V_WMMA_F32_16X16X4_F32,
// K-loop of 32 steps (K=128, 4 per WMMA). Block = 8 waves = 8 N-tiles.
__global__ void __launch_bounds__(256)
gcn_gemm_relu(const float* __restrict__ sums,
              const float* __restrict__ cnt,
              const float* __restrict__ W,
              const float* __restrict__ b,
              float* __restrict__ out) {
    const int m0   = blockIdx.x * 16;
    const int wave = threadIdx.x >> 5;   // 0..7 -> N tile
    const int lane = threadIdx.x & 31;
    const int n0   = wave * 16;

    const int arow  = m0 + (lane & 15);          // A-matrix row for this lane
    const int brow  = n0 + (lane & 15);          // W row (= output column)
    const int khalf = (lane >> 4) * 2;           // lanes 16..31 hold K+2,K+3

    // fused segment-mean: per-lane row scale, constant across the K loop
    const float inv = 1.0f / fmaxf(cnt[arow], 1.0f);

    const float* aptr = sums + (long long)arow * DIM + khalf;
    const float* bptr = W    + (long long)brow * DIM + khalf;

    v8f acc = {};
#pragma unroll 8
    for (int k0 = 0; k0 < DIM; k0 += 4) {
        v2f a = *(const v2f*)(aptr + k0);
        a.x *= inv;
        a.y *= inv;
        v2f bb = *(const v2f*)(bptr + k0);
        // 8 args: (neg_a, A, neg_b, B, c_mod, C, reuse_a, reuse_b)
        acc = __builtin_amdgcn_wmma_f32_16x16x4_f32(
            false, a, false, bb, (short)0, acc, false, false);
    }

    // epilogue: bias + ReLU, coalesced stores.
    // D layout: VGPR j, lanes 0-15 -> M=m0+j, lanes 16-31 -> M=m0+j+8; N=n0+(lane&15)
    const float bias  = b[brow];
    const int   mbase = m0 + ((lane >> 4) << 3);
    const int   coln  = n0 + (lane & 15);
#pragma unroll
    for (int j = 0; j < 8; ++j) {
        float v = acc[j] + bias;
        out[(long long)(mbase + j) * DIM + coln] = fmaxf(v, 0.0f);
    }
}

// ------------------------------------------------------------------- launcher
extern "C" void kernel_launch(void* const* d_in, const int* in_sizes, int n_in,
                              void* d_out, int out_size, void* d_ws, size_t ws_size,
                              hipStream_t stream) {
    const float* feature = (const float*)d_in[0];   // [N, D] f32
    const int*   src     = (const int*)  d_in[1];   // [E]
    const int*   dst     = (const int*)  d_in[2];   // [E]
    const float* W       = (const float*)d_in[3];   // [D, D] f32
    const float* b       = (const float*)d_in[4];   // [D]    f32
    float*       out     = (float*)d_out;           // [N, D] f32

    float* sums = (float*)d_ws;                     // N*D floats
    float* cnt  = sums + (size_t)NNODES * DIM;      // N floats

    // 1) zero accumulators (must re-zero every call: graph replay, poisoned ws)
    {
        int total  = NNODES * DIM + NNODES;
        int blocks = (total + 255) / 256;
        gcn_zero_ws<<<blocks, 256, 0, stream>>>((float*)d_ws, total);
    }
    // 2) edge gather + scatter-add (wave per edge; 8 edges per 256-thread block)
    {
        int blocks = (NEDGES + 7) / 8;
        gcn_scatter<<<blocks, 256, 0, stream>>>(feature, src, dst, sums, cnt);
    }
    // 3) fused mean + GEMM(WMMA f32) + bias + ReLU
    {
        int blocks = NNODES / 16;                   // 3125 M-tiles, 8 N-tiles/block
        gcn_gemm_relu<<<blocks, 256, 0, stream>>>(sums, cnt, W, b, out);
    }
    (void)in_sizes; (void)n_in; (void)out_size; (void)ws_size;
}